// Model_86242943303990
// MI455X (gfx1250) — compile-verified
//
#include <hip/hip_runtime.h>
#include <math.h>

typedef __attribute__((ext_vector_type(2))) float v2f;
typedef __attribute__((ext_vector_type(8))) float v8f;

#define NPTS    10000
#define NTILES  625          // NPTS / 16
#define DEG2RAD 0.017453292519943295f

// ---------------------------------------------------------------------------
// 1) p = (verts + t) @ (Rz*Ry*Rx)^T   (each thread recomputes the tiny R)
// ---------------------------------------------------------------------------
__global__ void transform_kernel(const float* __restrict__ verts,
                                 const float* __restrict__ trans,
                                 const float* __restrict__ ax,
                                 const float* __restrict__ ay,
                                 const float* __restrict__ az,
                                 float* __restrict__ out, int V)
{
    int i = blockIdx.x * blockDim.x + threadIdx.x;
    if (i >= V) return;
    float tx = ax[0] * DEG2RAD, ty = ay[0] * DEG2RAD, tz = az[0] * DEG2RAD;
    float cx = cosf(tx), sx = sinf(tx);
    float cy = cosf(ty), sy = sinf(ty);
    float cz = cosf(tz), sz = sinf(tz);
    // M = Rz * Ry * Rx
    float m00 = cz * cy;
    float m01 = cz * sy * sx - sz * cx;
    float m02 = cz * sy * cx + sz * sx;
    float m10 = sz * cy;
    float m11 = sz * sy * sx + cz * cx;
    float m12 = sz * sy * cx - cz * sx;
    float m20 = -sy;
    float m21 = cy * sx;
    float m22 = cy * cx;

    float q0 = verts[i * 3 + 0] + trans[0];
    float q1 = verts[i * 3 + 1] + trans[1];
    float q2 = verts[i * 3 + 2] + trans[2];
    out[i * 3 + 0] = m00 * q0 + m01 * q1 + m02 * q2;
    out[i * 3 + 1] = m10 * q0 + m11 * q1 + m12 * q2;
    out[i * 3 + 2] = m20 * q0 + m21 * q1 + m22 * q2;
}

// ---------------------------------------------------------------------------
// 2) Barycentric surface sampling + face normals. Writes points/normals with
//    stride 4 (x,y,z,0) so the chamfer kernel can do aligned 8B lane loads,
//    plus |p|^2 per point.
// ---------------------------------------------------------------------------
__global__ void sample_kernel(const float* __restrict__ verts,
                              const int*  __restrict__ faces,
                              const int*  __restrict__ fidx,
                              const float* __restrict__ uv,
                              float* __restrict__ pts4,
                              float* __restrict__ n4,
                              float* __restrict__ xx, int N)
{
    int i = blockIdx.x * blockDim.x + threadIdx.x;
    if (i >= N) return;
    int f  = fidx[i];
    int a0 = faces[f * 3 + 0], a1 = faces[f * 3 + 1], a2 = faces[f * 3 + 2];
    float v0x = verts[a0 * 3 + 0], v0y = verts[a0 * 3 + 1], v0z = verts[a0 * 3 + 2];
    float v1x = verts[a1 * 3 + 0], v1y = verts[a1 * 3 + 1], v1z = verts[a1 * 3 + 2];
    float v2x = verts[a2 * 3 + 0], v2y = verts[a2 * 3 + 1], v2z = verts[a2 * 3 + 2];

    float su = sqrtf(uv[i * 2 + 0]);
    float vv = uv[i * 2 + 1];
    float w0 = 1.0f - su, w1 = su * (1.0f - vv), w2 = su * vv;

    float px = w0 * v0x + w1 * v1x + w2 * v2x;
    float py = w0 * v0y + w1 * v1y + w2 * v2y;
    float pz = w0 * v0z + w1 * v1z + w2 * v2z;

    float e1x = v1x - v0x, e1y = v1y - v0y, e1z = v1z - v0z;
    float e2x = v2x - v0x, e2y = v2y - v0y, e2z = v2z - v0z;
    float nx = e1y * e2z - e1z * e2y;
    float ny = e1z * e2x - e1x * e2z;
    float nz = e1x * e2y - e1y * e2x;
    float inv = 1.0f / (sqrtf(nx * nx + ny * ny + nz * nz) + 1e-8f);

    pts4[i * 4 + 0] = px; pts4[i * 4 + 1] = py;
    pts4[i * 4 + 2] = pz; pts4[i * 4 + 3] = 0.0f;     // pad -> K=4 zero slot
    n4[i * 4 + 0] = nx * inv; n4[i * 4 + 1] = ny * inv;
    n4[i * 4 + 2] = nz * inv; n4[i * 4 + 3] = 0.0f;
    xx[i] = px * px + py * py + pz * pz;
}

// ---------------------------------------------------------------------------
// 3) One-direction chamfer: for each row tile (16 points of X) scan all 625
//    column tiles of Y with V_WMMA_F32_16X16X4_F32, fuse d2 = xx+yy-2*S,
//    keep running (min,argmin) per row, reduce across lanes/waves, add the
//    normal term, write per-row loss.
// ---------------------------------------------------------------------------
__global__ __launch_bounds__(256)
void chamfer_dir_kernel(const float* __restrict__ X4,  const float* __restrict__ Y4,
                        const float* __restrict__ XX,  const float* __restrict__ YY,
                        const float* __restrict__ NX4, const float* __restrict__ NY4,
                        float* __restrict__ rowloss)
{
    __shared__ float s_d2[8][16];
    __shared__ int   s_j[8][16];

    const int wave = threadIdx.x >> 5;   // 8 waves per block
    const int lane = threadIdx.x & 31;
    const int m    = lane & 15;          // column within tile / row for A load
    const int half = lane >> 4;          // 0: (x,y) / K=0,1   1: (z,0) / K=2,3
    const int i0   = blockIdx.x * 16;    // row tile base

    // A-matrix 16x4 f32 layout: lanes 0-15 hold K=0,1; lanes 16-31 hold K=2,3
    v2f a = *(const v2f*)(X4 + (size_t)(i0 + m) * 4 + half * 2);

    float xrow[8];
#pragma unroll
    for (int v = 0; v < 8; ++v) xrow[v] = XX[i0 + v + half * 8];

    float best[8];
    int   bestj[8];
#pragma unroll
    for (int v = 0; v < 8; ++v) { best[v] = 3.4e38f; bestj[v] = 0x7fffffff; }

    for (int ct = wave; ct < NTILES; ct += 8) {     // uniform per wave
        const int j0 = ct * 16;
        // B-matrix 4x16: column n = point j0+n; same lane layout as A
        v2f b = *(const v2f*)(Y4 + (size_t)(j0 + m) * 4 + half * 2);
        float yyn = YY[j0 + m];
        v8f c = {};
        c = __builtin_amdgcn_wmma_f32_16x16x4_f32(
                /*neg_a=*/false, a, /*neg_b=*/false, b,
                /*c_mod=*/(short)0, c, /*reuse_a=*/false, /*reuse_b=*/false);
#pragma unroll
        for (int v = 0; v < 8; ++v) {
            // D layout: VGPR v holds row M = v + 8*half, lane column N = m
            float cand = xrow[v] + yyn - 2.0f * c[v];
            int   jc   = j0 + m;
            if (cand < best[v] || (cand == best[v] && jc < bestj[v])) {
                best[v] = cand; bestj[v] = jc;
            }
        }
    }

    // butterfly min+argmin inside each 16-lane half (rows v and v+8)
#pragma unroll
    for (int v = 0; v < 8; ++v) {
        float bd = best[v];
        int   bj = bestj[v];
#pragma unroll
        for (int off = 1; off <= 8; off <<= 1) {
            float od = __shfl_xor(bd, off, 32);
            int   oj = __shfl_xor(bj, off, 32);
            if (od < bd || (od == bd && oj < bj)) { bd = od; bj = oj; }
        }
        if (m == 0) {                     // lanes 0 and 16
            s_d2[wave][v + half * 8] = bd;
            s_j [wave][v + half * 8] = bj;
        }
    }
    __syncthreads();

    if (threadIdx.x < 16) {
        int   r  = threadIdx.x;
        float bd = s_d2[0][r];
        int   bj = s_j [0][r];
#pragma unroll
        for (int w = 1; w < 8; ++w) {
            float od = s_d2[w][r];
            int   oj = s_j [w][r];
            if (od < bd || (od == bd && oj < bj)) { bd = od; bj = oj; }
        }
        const float* nx = NX4 + (size_t)(i0 + r) * 4;
        const float* ny = NY4 + (size_t)bj * 4;
        float dp = nx[0] * ny[0] + nx[1] * ny[1] + nx[2] * ny[2];
        rowloss[i0 + r] = bd + (1.0f - fabsf(dp));
    }
}

// ---------------------------------------------------------------------------
// 4) Deterministic fixed-order reduction of both per-row loss arrays.
// ---------------------------------------------------------------------------
__global__ void reduce_kernel(const float* __restrict__ rlx,
                              const float* __restrict__ rly,
                              float* __restrict__ out)
{
    __shared__ float s[256];
    float acc = 0.0f;
    for (int i = threadIdx.x; i < NPTS; i += 256) acc += rlx[i] + rly[i];
    s[threadIdx.x] = acc;
    __syncthreads();
    for (int off = 128; off > 0; off >>= 1) {
        if (threadIdx.x < off) s[threadIdx.x] += s[threadIdx.x + off];
        __syncthreads();
    }
    if (threadIdx.x == 0) out[0] = s[0] / (float)NPTS;
}

// ---------------------------------------------------------------------------
extern "C" void kernel_launch(void* const* d_in, const int* in_sizes, int n_in,
                              void* d_out, int out_size, void* d_ws, size_t ws_size,
                              hipStream_t stream)
{
    const float* verts_mov = (const float*)d_in[0];
    const int*   faces_mov = (const int*)  d_in[1];
    const float* verts_fix = (const float*)d_in[2];
    const int*   faces_fix = (const int*)  d_in[3];
    const float* trans     = (const float*)d_in[4];
    const float* angle_x   = (const float*)d_in[5];
    const float* angle_y   = (const float*)d_in[6];
    const float* angle_z   = (const float*)d_in[7];
    const int*   fidx_mov  = (const int*)  d_in[8];
    const float* uv_mov    = (const float*)d_in[9];
    const int*   fidx_fix  = (const int*)  d_in[10];
    const float* uv_fix    = (const float*)d_in[11];

    const int V = in_sizes[0] / 3;      // 40000

    float* out   = (float*)d_out;       // [0] = loss, [1..] = p (V x 3)
    float* p_out = out + 1;

    float* ws        = (float*)d_ws;
    float* pts_m4    = ws;                      // NPTS*4
    float* n_m4      = ws + 1 * NPTS * 4;
    float* pts_f4    = ws + 2 * NPTS * 4;
    float* n_f4      = ws + 3 * NPTS * 4;
    float* xx_m      = ws + 4 * NPTS * 4;       // NPTS
    float* xx_f      = xx_m + NPTS;
    float* rowloss_x = xx_f + NPTS;             // NPTS
    float* rowloss_y = rowloss_x + NPTS;        // NPTS

    transform_kernel<<<(V + 255) / 256, 256, 0, stream>>>(
        verts_mov, trans, angle_x, angle_y, angle_z, p_out, V);

    sample_kernel<<<(NPTS + 255) / 256, 256, 0, stream>>>(
        p_out, faces_mov, fidx_mov, uv_mov, pts_m4, n_m4, xx_m, NPTS);
    sample_kernel<<<(NPTS + 255) / 256, 256, 0, stream>>>(
        verts_fix, faces_fix, fidx_fix, uv_fix, pts_f4, n_f4, xx_f, NPTS);

    chamfer_dir_kernel<<<NTILES, 256, 0, stream>>>(
        pts_m4, pts_f4, xx_m, xx_f, n_m4, n_f4, rowloss_x);
    chamfer_dir_kernel<<<NTILES, 256, 0, stream>>>(
        pts_f4, pts_m4, xx_f, xx_m, n_f4, n_m4, rowloss_y);

    reduce_kernel<<<1, 256, 0, stream>>>(rowloss_x, rowloss_y, out);
}